// GCN_29618094473880
// MI455X (gfx1250) — compile-verified
//
#include <hip/hip_runtime.h>
#include <hip/hip_bf16.h>
#include <math.h>

typedef float v2f __attribute__((ext_vector_type(2)));
typedef float v8f __attribute__((ext_vector_type(8)));

constexpr int Nn  = 100000;   // nodes
constexpr int Ee  = 3200000;  // edges
constexpr int INC = 512;      // in channels
constexpr int HID = 16;       // hidden == out channels

// ---------------------------------------------------------------- init
__global__ void k_init(float* __restrict__ deg,
                       float* __restrict__ agg1,
                       float* __restrict__ agg2) {
  int i = blockIdx.x * blockDim.x + threadIdx.x;
  if (i < Nn * HID) { agg1[i] = 0.0f; agg2[i] = 0.0f; }
  if (i < Nn) deg[i] = 1.0f;  // self-loop contribution to degree
}

__device__ __forceinline__ void atomAddF(float* p, float v) {
  unsafeAtomicAdd(p, v);  // hardware global_atomic_add_f32
}

// ---------------------------------------------------------------- degree
__global__ void k_degree(const long long* __restrict__ ei,
                         float* __restrict__ deg) {
  int e = blockIdx.x * blockDim.x + threadIdx.x;
  if (e >= Ee) return;
  int d = (int)ei[Ee + e];  // dst row of edge_index [2, E]
  atomAddF(&deg[d], 1.0f);
}

__global__ void k_dinv(float* __restrict__ deg) {
  int i = blockIdx.x * blockDim.x + threadIdx.x;
  if (i < Nn) deg[i] = rsqrtf(deg[i]);  // deg -> deg^{-1/2} in place
}

// ---------------------------------------------------------------- WMMA GEMM
// out[rows, 16] = A[rows, K] @ W[K, 16] using V_WMMA_F32_16X16X4_F32.
// One wave per 16-row tile. W staged in LDS transposed (+pad) so B fetch
// is a single ds_load_b64 per k-step.
#define GEMM_WAVES 8
__global__ void __launch_bounds__(256) k_gemm_wmma(
    const float* __restrict__ A, const float* __restrict__ W,
    float* __restrict__ out, int K, int nrows) {
  __shared__ float wT[16 * (512 + 2)];
  const int stride = K + 2;
  for (int i = threadIdx.x; i < K * 16; i += blockDim.x) {
    int k = i >> 4, n = i & 15;
    wT[n * stride + k] = W[i];  // wT[n][k] = W[k][n]
  }
  __syncthreads();

  const int wave = threadIdx.x >> 5;
  const int lane = threadIdx.x & 31;
  const long long rowBase = ((long long)blockIdx.x * GEMM_WAVES + wave) * 16;
  if (rowBase >= nrows) return;  // wave-uniform guard: EXEC stays all-ones

  const int m  = lane & 15;  // A-row in tile; also D-column n
  const int kh = lane >> 4;  // which k-pair this half-wave holds
  const float* __restrict__ aRow = A + (rowBase + m) * (long long)K + 2 * kh;
  const float* __restrict__ bRow = wT + m * stride + 2 * kh;

  v8f acc = {};
#pragma unroll 8
  for (int kk = 0; kk < K; kk += 4) {
    v2f a = *reinterpret_cast<const v2f*>(aRow + kk);  // A(m, k0), A(m, k0+1)
    v2f b = *reinterpret_cast<const v2f*>(bRow + kk);  // B(k0, n), B(k0+1, n)
    acc = __builtin_amdgcn_wmma_f32_16x16x4_f32(
        /*neg_a=*/false, a, /*neg_b=*/false, b,
        /*c_mod=*/(short)0, acc, /*reuse_a=*/false, /*reuse_b=*/false);
  }

  float* __restrict__ oBase = out + rowBase * HID;
#pragma unroll
  for (int v = 0; v < 8; ++v) {
    int rm = v + 8 * kh;           // D row
    oBase[rm * HID + m] = acc[v];  // D(m=rm, n=m)
  }
}

// ---------------------------------------------------------------- scatter
// agg[dst,c] += h[src,c] * dinv[src]*dinv[dst]; 16 lanes = 16 channels of one
// edge -> coalesced gathers and cacheline-coalesced atomics (L2-resident).
__global__ void k_scatter(const long long* __restrict__ ei,
                          const float* __restrict__ h,
                          const float* __restrict__ dinv,
                          float* __restrict__ agg) {
  long long t = (long long)blockIdx.x * blockDim.x + threadIdx.x;
  long long e = t >> 4;
  int c = (int)(t & 15);
  if (e >= Ee) return;
  int s = (int)ei[e];
  int d = (int)ei[Ee + e];
  float nrm = dinv[s] * dinv[d];
  atomAddF(&agg[(long long)d * HID + c], h[(long long)s * HID + c] * nrm);
}

// ---------------------------------------------------------------- layer-1 post
// agg1 <- relu(agg1 + h1/deg + b1) * drop_mask   (in place; becomes gemm2 input)
__global__ void k_post1(float* __restrict__ agg1, const float* __restrict__ h1,
                        const float* __restrict__ dinv,
                        const float* __restrict__ b1,
                        const float* __restrict__ drop) {
  int i = blockIdx.x * blockDim.x + threadIdx.x;
  if (i >= Nn * HID) return;
  int n = i >> 4, c = i & 15;
  float di = dinv[n];
  float v = agg1[i] + h1[i] * di * di + b1[c];
  agg1[i] = fmaxf(v, 0.0f) * drop[i];
}

// ---------------------------------------------------------------- final
// out = log_softmax(agg2 + h2/deg + b2, axis=1)
__global__ void k_final(const float* __restrict__ agg2,
                        const float* __restrict__ h2,
                        const float* __restrict__ dinv,
                        const float* __restrict__ b2,
                        float* __restrict__ out) {
  int i = blockIdx.x * blockDim.x + threadIdx.x;
  if (i >= Nn) return;
  float di = dinv[i];
  float invd = di * di;
  const long long base = (long long)i * 16;
  float v[16];
  float mx = -1e30f;
#pragma unroll
  for (int c = 0; c < 16; ++c) {
    v[c] = agg2[base + c] + h2[base + c] * invd + b2[c];
    mx = fmaxf(mx, v[c]);
  }
  float s = 0.0f;
#pragma unroll
  for (int c = 0; c < 16; ++c) s += __expf(v[c] - mx);
  float lse = __logf(s);
#pragma unroll
  for (int c = 0; c < 16; ++c) out[base + c] = v[c] - mx - lse;
}

// ---------------------------------------------------------------- launch
extern "C" void kernel_launch(void* const* d_in, const int* in_sizes, int n_in,
                              void* d_out, int out_size, void* d_ws,
                              size_t ws_size, hipStream_t stream) {
  (void)in_sizes; (void)n_in; (void)out_size; (void)ws_size;
  const float*     x    = (const float*)d_in[0];
  const long long* ei   = (const long long*)d_in[1];  // int64 [2, E]
  const float*     W1   = (const float*)d_in[2];
  const float*     b1   = (const float*)d_in[3];
  const float*     W2   = (const float*)d_in[4];
  const float*     b2   = (const float*)d_in[5];
  const float*     drop = (const float*)d_in[6];
  float*           out  = (float*)d_out;

  // workspace carve-up (~26 MB)
  float* f    = (float*)d_ws;
  float* deg  = f;                          // Nn floats (becomes dinv)
  float* h1   = f + 131072;                 // Nn*16
  float* agg1 = h1 + (size_t)Nn * HID;      // Nn*16 (becomes dropped hidden)
  float* h2   = agg1 + (size_t)Nn * HID;    // Nn*16
  float* agg2 = h2 + (size_t)Nn * HID;      // Nn*16

  const int T = 256;
  const int gInit    = (Nn * HID + T - 1) / T;
  const int gEdge    = (Ee + T - 1) / T;
  const int gNode    = (Nn + T - 1) / T;
  const int gScatter = (int)(((long long)Ee * 16 + T - 1) / T);
  const int gGemm    = ((Nn / 16) + GEMM_WAVES - 1) / GEMM_WAVES;

  k_init<<<gInit, T, 0, stream>>>(deg, agg1, agg2);
  k_degree<<<gEdge, T, 0, stream>>>(ei, deg);
  k_dinv<<<gNode, T, 0, stream>>>(deg);

  // layer 1
  k_gemm_wmma<<<gGemm, T, 0, stream>>>(x, W1, h1, INC, Nn);
  k_scatter<<<gScatter, T, 0, stream>>>(ei, h1, deg, agg1);
  k_post1<<<gInit, T, 0, stream>>>(agg1, h1, deg, b1, drop);

  // layer 2
  k_gemm_wmma<<<gGemm, T, 0, stream>>>(agg1, W2, h2, HID, Nn);
  k_scatter<<<gScatter, T, 0, stream>>>(ei, h2, deg, agg2);
  k_final<<<gNode, T, 0, stream>>>(agg2, h2, deg, b2, out);
}